// SDNetLoss_16698832846993
// MI455X (gfx1250) — compile-verified
//
#include <hip/hip_runtime.h>
#include <hip/hip_bf16.h>

#define CNT      8192
#define DIM      128
#define NMAT     3
#define NDOT     (6 * CNT)          // 49152 dots for alpha estimation
#define MARGIN_F 0.36f
#define EPS_SIM  0.01f
#define BIG_F    99999.0f

#define BM    128                   // rows per block (8 waves x 16)
#define BN    64                    // cols per block (4 wmma tiles)
#define KC    64                    // K chunk staged in LDS (DIM = 2 chunks)
#define LDSTR 68                    // padded LDS row stride (floats)

typedef __attribute__((ext_vector_type(2))) float v2f;
typedef __attribute__((ext_vector_type(8))) float v8f;

// monotonic float->uint map: a<b  <=>  ord(a)<ord(b)
__device__ __forceinline__ unsigned int ordered_f32(float f) {
  unsigned int u = __float_as_uint(f);
  return (u & 0x80000000u) ? ~u : (u | 0x80000000u);
}

__device__ __forceinline__ float clip1(float d) {
  return fminf(1.0f, fmaxf(-1.0f, d));
}

__device__ __forceinline__ float softplusf(float x) {
  if (x > 20.0f)  return x;
  if (x < -20.0f) return expf(x);
  return log1pf(expf(x));
}

// dot_to_dist
__device__ __forceinline__ float d2d(float dot, float alpha) {
  float om = 1.0f - dot;
  return alpha * om + (1.0f - alpha) * sqrtf(2.0f * om + 1e-12f);
}

__device__ __forceinline__ int a_base_of(int z) { return (z == 0) ? 0 : (z == 1 ? 2 * CNT : 4 * CNT); }
__device__ __forceinline__ int b_base_of(int z) { return (z == 0) ? CNT : (z == 1 ? 5 * CNT : 3 * CNT); }

// ---------------------------------------------------------------- init packed
__global__ void init_kernel(unsigned long long* __restrict__ packed) {
  int i = blockIdx.x * blockDim.x + threadIdx.x;
  if (i < NMAT * 2 * CNT) packed[i] = 0xFFFFFFFFFFFFFFFFull;
}

// ------------------------------------------------------- reciprocal row norms
__global__ void rnorm_kernel(const float* __restrict__ descs, float* __restrict__ rnorm) {
  int wave = (int)((blockIdx.x * blockDim.x + threadIdx.x) >> 5);
  int lane = threadIdx.x & 31;
  if (wave >= 6 * CNT) return;
  float4 v = *(const float4*)(descs + (size_t)wave * DIM + lane * 4);
  float s = v.x * v.x + v.y * v.y + v.z * v.z + v.w * v.w;
  #pragma unroll
  for (int m = 16; m >= 1; m >>= 1) s += __shfl_xor(s, m);
  if (lane == 0) rnorm[wave] = 1.0f / sqrtf(s);
}

// ------------------------------------- pre-normalized panel for async copies
__global__ void xn_kernel(const float* __restrict__ descs,
                          const float* __restrict__ rnorm,
                          float* __restrict__ xn) {
  size_t i = (size_t)blockIdx.x * blockDim.x + threadIdx.x;  // one float4 each
  size_t row = i >> 5;                                       // 32 float4 / row
  int c4 = (int)(i & 31) << 2;
  float rn = rnorm[row];
  float4 v = *(const float4*)(descs + row * DIM + c4);
  float4 o; o.x = v.x * rn; o.y = v.y * rn; o.z = v.z * rn; o.w = v.w * rn;
  *(float4*)(xn + row * DIM + c4) = o;
}

// ----------------------------------------------------------- shared epilogue
// C layout: acc[t][v] lives at row = v + 8*(lane>>4), col = t*16 + (lane&15)
__device__ __forceinline__ void minmine_epilogue(
    const v8f acc[4], int mBase, int nBase, int w, int lane, int tid,
    unsigned long long* rowmin_g, unsigned long long* colmin_g,
    unsigned long long* colmin_s) {
  unsigned int ordv[4][8];
  #pragma unroll
  for (int t = 0; t < 4; ++t) {
    #pragma unroll
    for (int v = 0; v < 8; ++v) {
      float s = acc[t][v];
      float d = 1.0f / s + EPS_SIM;
      int gr = mBase + (w << 4) + v + ((lane >> 4) << 3);
      int gc = nBase + (t << 4) + (lane & 15);
      if (gr == gc) d += BIG_F;
      ordv[t][v] = ordered_f32(d);
    }
  }
  const int rlo = mBase + (w << 4) + ((lane >> 4) << 3);
  const int cl  = nBase + (lane & 15);

  // row minima: reduce over cols (4 tiles in-lane, then 16 lanes per half)
  #pragma unroll
  for (int v = 0; v < 8; ++v) {
    unsigned long long pr = 0xFFFFFFFFFFFFFFFFull;
    #pragma unroll
    for (int t = 0; t < 4; ++t) {
      unsigned long long p =
          ((unsigned long long)ordv[t][v] << 32) | (unsigned int)(cl + (t << 4));
      pr = (p < pr) ? p : pr;
    }
    #pragma unroll
    for (int m = 1; m < 16; m <<= 1) {
      unsigned long long o = __shfl_xor(pr, m);
      pr = (o < pr) ? o : pr;
    }
    if ((lane & 15) == 0) atomicMin(&rowmin_g[rlo + v], pr);
  }

  // col minima: reduce over rows (8 in-lane, combine halves, then LDS u64 min)
  #pragma unroll
  for (int t = 0; t < 4; ++t) {
    unsigned long long pc = 0xFFFFFFFFFFFFFFFFull;
    #pragma unroll
    for (int v = 0; v < 8; ++v) {
      unsigned long long p =
          ((unsigned long long)ordv[t][v] << 32) | (unsigned int)(rlo + v);
      pc = (p < pc) ? p : pc;
    }
    unsigned long long o = __shfl_xor(pc, 16);
    pc = (o < pc) ? o : pc;
    if (lane < 16) atomicMin(&colmin_s[(t << 4) + lane], pc);  // ds_min_u64
  }
  __syncthreads();
  if (tid < BN) atomicMin(&colmin_g[nBase + tid], colmin_s[tid]);
}

// ----------------------------------------------------- WMMA compute (shared)
__device__ __forceinline__ void wmma_chunk(const float* As, const float* Bs,
                                           int w, int lane, v8f acc[4]) {
  const int arow     = (w << 4) + (lane & 15);
  const int koffLane = (lane >> 4) << 1;
  #pragma unroll
  for (int k4 = 0; k4 < KC / 4; ++k4) {
    int kk = (k4 << 2) + koffLane;
    v2f afrag = *(const v2f*)&As[arow * LDSTR + kk];
    #pragma unroll
    for (int t = 0; t < 4; ++t) {
      v2f bfrag = *(const v2f*)&Bs[((t << 4) + (lane & 15)) * LDSTR + kk];
      acc[t] = __builtin_amdgcn_wmma_f32_16x16x4_f32(
          false, afrag, false, bfrag, (short)0, acc[t], false, false);
    }
  }
}

// ------------------- fused GEMM + row/col min/argmin, ASYNC-to-LDS staging
// Panels come from the pre-normalized xn buffer; global->LDS copies use
// GLOBAL_LOAD_ASYNC_TO_LDS_B128 (ASYNCcnt), bypassing VGPRs entirely.
__global__ __launch_bounds__(256)
void gemm_min_async_kernel(const float* __restrict__ xn,
                           unsigned long long* __restrict__ packed) {
  __shared__ float As[BM * LDSTR];
  __shared__ float Bs[BN * LDSTR];
  __shared__ unsigned long long colmin_s[BN];

  const int z     = blockIdx.z;
  const int aBase = a_base_of(z);
  const int bBase = b_base_of(z);
  const int mBase = blockIdx.y * BM;
  const int nBase = blockIdx.x * BN;
  const int tid   = threadIdx.x;
  const int lane  = tid & 31;
  const int w     = tid >> 5;

  unsigned long long* rowmin_g = packed + (size_t)z * 2 * CNT;
  unsigned long long* colmin_g = rowmin_g + CNT;

  if (tid < BN) colmin_s[tid] = 0xFFFFFFFFFFFFFFFFull;

  v8f acc[4] = {};

  for (int kc = 0; kc < DIM; kc += KC) {
    __syncthreads();  // previous chunk's consumers done with LDS
    // A panel: 128 rows x 64 floats -> 8 b128 async copies per thread
    for (int idx = tid; idx < BM * (KC / 4); idx += 256) {
      int row = idx >> 4;
      int c4  = (idx & 15) << 2;
      unsigned int ldsa = (unsigned int)(uintptr_t)&As[row * LDSTR + c4];
      const float* g = xn + (size_t)(aBase + mBase + row) * DIM + kc + c4;
      if (kc == 0) __builtin_prefetch(g + KC, 0, 0);   // warm next chunk in L2
      asm volatile("global_load_async_to_lds_b128 %0, %1, off"
                   :: "v"(ldsa), "v"((unsigned long long)(uintptr_t)g)
                   : "memory");
    }
    // B panel: 64 rows x 64 floats -> 4 b128 async copies per thread
    for (int idx = tid; idx < BN * (KC / 4); idx += 256) {
      int row = idx >> 4;
      int c4  = (idx & 15) << 2;
      unsigned int ldsa = (unsigned int)(uintptr_t)&Bs[row * LDSTR + c4];
      const float* g = xn + (size_t)(bBase + nBase + row) * DIM + kc + c4;
      if (kc == 0) __builtin_prefetch(g + KC, 0, 0);
      asm volatile("global_load_async_to_lds_b128 %0, %1, off"
                   :: "v"(ldsa), "v"((unsigned long long)(uintptr_t)g)
                   : "memory");
    }
    asm volatile("s_wait_asynccnt 0x0" ::: "memory");
    __syncthreads();
    wmma_chunk(As, Bs, w, lane, acc);
  }

  minmine_epilogue(acc, mBase, nBase, w, lane, tid, rowmin_g, colmin_g, colmin_s);
}

// ---------------- fallback (small workspace): stage + normalize via VGPRs
__global__ __launch_bounds__(256)
void gemm_min_kernel(const float* __restrict__ descs,
                     const float* __restrict__ rnorm,
                     unsigned long long* __restrict__ packed) {
  __shared__ float As[BM * LDSTR];
  __shared__ float Bs[BN * LDSTR];
  __shared__ unsigned long long colmin_s[BN];

  const int z     = blockIdx.z;
  const int aBase = a_base_of(z);
  const int bBase = b_base_of(z);
  const int mBase = blockIdx.y * BM;
  const int nBase = blockIdx.x * BN;
  const int tid   = threadIdx.x;
  const int lane  = tid & 31;
  const int w     = tid >> 5;

  unsigned long long* rowmin_g = packed + (size_t)z * 2 * CNT;
  unsigned long long* colmin_g = rowmin_g + CNT;

  if (tid < BN) colmin_s[tid] = 0xFFFFFFFFFFFFFFFFull;

  v8f acc[4] = {};

  for (int kc = 0; kc < DIM; kc += KC) {
    __syncthreads();
    for (int idx = tid; idx < BM * (KC / 4); idx += 256) {
      int row = idx >> 4;
      int c4  = (idx & 15) << 2;
      int gr  = aBase + mBase + row;
      const float* src = descs + (size_t)gr * DIM + kc + c4;
      if (kc == 0) __builtin_prefetch(src + KC, 0, 0);
      float rn = rnorm[gr];
      float4 v = *(const float4*)src;
      float4 o; o.x = v.x * rn; o.y = v.y * rn; o.z = v.z * rn; o.w = v.w * rn;
      *(float4*)&As[row * LDSTR + c4] = o;
    }
    for (int idx = tid; idx < BN * (KC / 4); idx += 256) {
      int row = idx >> 4;
      int c4  = (idx & 15) << 2;
      int gr  = bBase + nBase + row;
      const float* src = descs + (size_t)gr * DIM + kc + c4;
      if (kc == 0) __builtin_prefetch(src + KC, 0, 0);
      float rn = rnorm[gr];
      float4 v = *(const float4*)src;
      float4 o; o.x = v.x * rn; o.y = v.y * rn; o.z = v.z * rn; o.w = v.w * rn;
      *(float4*)&Bs[row * LDSTR + c4] = o;
    }
    __syncthreads();
    wmma_chunk(As, Bs, w, lane, acc);
  }

  minmine_epilogue(acc, mBase, nBase, w, lane, tid, rowmin_g, colmin_g, colmin_s);
}

// --------------------------------- pos (diag) and neg (mined) dots, 1 wave ea
__global__ void posneg_kernel(const float* __restrict__ descs,
                              const float* __restrict__ rnorm,
                              const unsigned long long* __restrict__ packed,
                              float* __restrict__ dots) {
  int gw   = (int)((blockIdx.x * blockDim.x + threadIdx.x) >> 5);
  int lane = threadIdx.x & 31;
  if (gw >= NMAT * CNT) return;
  int z = gw / CNT, i = gw - z * CNT;
  int aBase = a_base_of(z), bBase = b_base_of(z);
  const unsigned long long* rowmin = packed + (size_t)z * 2 * CNT;
  const unsigned long long* colmin = rowmin + CNT;

  size_t ra = (size_t)(aBase + i);
  float4 va = *(const float4*)(descs + ra * DIM + lane * 4);

  // positive: a_i . b_i
  size_t rp = (size_t)(bBase + i);
  float4 vp = *(const float4*)(descs + rp * DIM + lane * 4);
  float sp = va.x * vp.x + va.y * vp.y + va.z * vp.z + va.w * vp.w;
  #pragma unroll
  for (int m = 16; m >= 1; m >>= 1) sp += __shfl_xor(sp, m);
  if (lane == 0) dots[(size_t)z * CNT + i] = sp * rnorm[ra] * rnorm[rp];

  // negative: idx_n = row_min < col_min ? (b side, row_arg) : (a side, col_arg)
  unsigned long long prow = rowmin[i], pcol = colmin[i];
  size_t rneg = ((prow >> 32) < (pcol >> 32))
                    ? (size_t)(bBase + (unsigned int)(prow & 0xFFFFFFFFu))
                    : (size_t)(aBase + (unsigned int)(pcol & 0xFFFFFFFFu));
  float4 vn = *(const float4*)(descs + rneg * DIM + lane * 4);
  float sn = va.x * vn.x + va.y * vn.y + va.z * vn.z + va.w * vn.w;
  #pragma unroll
  for (int m = 16; m >= 1; m >>= 1) sn += __shfl_xor(sn, m);
  if (lane == 0) dots[(size_t)(3 * CNT) + (size_t)z * CNT + i] = sn * rnorm[ra] * rnorm[rneg];
}

// ------------------------ alpha line search: stats are linear/quadratic in a
// dist(a) = sq + a*b,  b = om - sq.  Need Msq, Mb, S_aa, S_ab, S_bb.
__global__ void alpha_stats_kernel(const float* __restrict__ dots, float* __restrict__ stats) {
  __shared__ double red[5][256];
  int tid = threadIdx.x;
  double s1 = 0, s2 = 0, s3 = 0, s4 = 0, s5 = 0;
  for (int j = tid; j < NDOT; j += 256) {
    float dot = dots[j];
    float om  = 1.0f - dot;
    float sq  = sqrtf(2.0f * om + 1e-12f);
    double b  = (double)om - (double)sq;
    s1 += sq; s2 += b; s3 += (double)sq * sq; s4 += (double)sq * b; s5 += b * b;
  }
  red[0][tid] = s1; red[1][tid] = s2; red[2][tid] = s3; red[3][tid] = s4; red[4][tid] = s5;
  __syncthreads();
  for (int s = 128; s > 0; s >>= 1) {
    if (tid < s)
      for (int q = 0; q < 5; ++q) red[q][tid] += red[q][tid + s];
    __syncthreads();
  }
  if (tid == 0) {
    double n = (double)NDOT;
    double Msq = red[0][0] / n, Mb = red[1][0] / n;
    stats[0] = (float)Msq;
    stats[1] = (float)Mb;
    stats[2] = (float)(red[2][0] - n * Msq * Msq);   // S_aa
    stats[3] = (float)(red[3][0] - n * Msq * Mb);    // S_ab
    stats[4] = (float)(red[4][0] - n * Mb * Mb);     // S_bb
  }
}

// one block per alpha candidate
__global__ void bce_kernel(const float* __restrict__ dots,
                           const float* __restrict__ stats,
                           float* __restrict__ bce) {
  __shared__ double red[256];
  int k = blockIdx.x, tid = threadIdx.x;
  float alpha = -1.0f + 0.01f * (float)k;
  float mean  = stats[0] + alpha * stats[1];
  float var   = (stats[2] + 2.0f * alpha * stats[3] + alpha * alpha * stats[4]) / (float)(NDOT - 1);
  float istd  = 1.0f / sqrtf(var);
  double sum = 0;
  for (int j = tid; j < NDOT; j += 256) {
    float dot  = dots[j];
    float om   = 1.0f - dot;
    float sq   = sqrtf(2.0f * om + 1e-12f);
    float dist = sq + alpha * (om - sq);
    float zz   = (dist - mean) * istd;
    // label=1 (first 3*CNT): -log(sigmoid(-z)) = softplus(z); label=0: softplus(-z)
    sum += (j < 3 * CNT) ? (double)softplusf(zz) : (double)softplusf(-zz);
  }
  red[tid] = sum;
  __syncthreads();
  for (int s = 128; s > 0; s >>= 1) {
    if (tid < s) red[tid] += red[tid + s];
    __syncthreads();
  }
  if (tid == 0) bce[k] = (float)(red[0] / (double)NDOT);
}

// ----------------------------------------------------------- final 8 scalars
__global__ void final_kernel(const float* __restrict__ dots,
                             const float* __restrict__ bce,
                             const float* __restrict__ l_a,
                             const float* __restrict__ l_p,
                             float* __restrict__ out) {
  __shared__ double red[6][256];
  __shared__ float s_alpha[2];
  int tid = threadIdx.x;
  if (tid == 0) {
    int best = 0; float bv = bce[0];
    for (int k = 1; k < 200; ++k) { float v = bce[k]; if (v < bv) { bv = v; best = k; } }
    float aopt = -1.0f + 0.01f * (float)best;
    s_alpha[1] = aopt;
    s_alpha[0] = aopt * (1.0f - 0.995f);   // ALPHA_INIT=0 momentum update
  }
  __syncthreads();
  float alpha = s_alpha[0];
  double t1 = 0, t2 = 0, t3 = 0, sla = 0, slp = 0, smask = 0;
  for (int i = tid; i < CNT; i += 256) {
    float pap   = dots[i];
    float parpa = dots[CNT + i];
    float paapr = dots[2 * CNT + i];
    float nap   = dots[3 * CNT + i];
    float narpa = dots[4 * CNT + i];
    float naapr = dots[5 * CNT + i];
    float m = (MARGIN_F + acosf(clip1(pap)) - acosf(clip1(nap)) > 0.0f) ? 1.0f : 0.0f;
    t1 += (double)((d2d(pap, alpha) - d2d(nap, alpha)) * m);
    t2 += (double)((l_p[i] * d2d(parpa, alpha) - d2d(narpa, alpha)) * m);
    t3 += (double)((l_a[i] * d2d(paapr, alpha) - d2d(naapr, alpha)) * m);
    sla += l_a[i]; slp += l_p[i]; smask += m;
  }
  red[0][tid] = t1; red[1][tid] = t2; red[2][tid] = t3;
  red[3][tid] = sla; red[4][tid] = slp; red[5][tid] = smask;
  __syncthreads();
  for (int s = 128; s > 0; s >>= 1) {
    if (tid < s)
      for (int q = 0; q < 6; ++q) red[q][tid] += red[q][tid + s];
    __syncthreads();
  }
  if (tid == 0) {
    float lc    = (float)(red[0][0] / (double)CNT);   // * LAMBDA_CLEAN(=1)
    float larpa = (float)(red[1][0] / (double)CNT);   // * LAMBDA_ADV(=1)
    float laapr = (float)(red[2][0] / (double)CNT);
    out[0] = (lc + larpa + laapr) / 3.0f;
    out[1] = lc;
    out[2] = larpa;
    out[3] = laapr;
    out[4] = (float)((red[3][0] / (double)CNT + red[4][0] / (double)CNT) * 0.5);
    out[5] = s_alpha[0];
    out[6] = s_alpha[1];
    out[7] = (float)(red[5][0] / (double)CNT);
  }
}

// ---------------------------------------------------------------------------
extern "C" void kernel_launch(void* const* d_in, const int* in_sizes, int n_in,
                              void* d_out, int out_size, void* d_ws, size_t ws_size,
                              hipStream_t stream) {
  const float* descs = (const float*)d_in[0];
  const float* l_a   = (const float*)d_in[1];
  const float* l_p   = (const float*)d_in[2];
  float* out = (float*)d_out;

  // workspace layout
  char* ws = (char*)d_ws;
  float* rnorm = (float*)ws;                                       //  6*CNT f32
  float* dots  = (float*)(ws + 196608);                            //  6*CNT f32
  unsigned long long* packed = (unsigned long long*)(ws + 393216); // 3*2*CNT u64
  float* stats = (float*)(ws + 786432);                            //  8 f32
  float* bce   = (float*)(ws + 786464);                            //  200 f32
  const size_t XN_OFF  = 787264;                                   // 16B aligned
  const size_t NEED_XN = XN_OFF + (size_t)6 * CNT * DIM * sizeof(float);
  float* xn = (float*)(ws + XN_OFF);                               // 6*CNT*DIM f32

  const bool big_ws = (ws_size >= NEED_XN);   // pure function of ws_size

  init_kernel<<<(NMAT * 2 * CNT + 255) / 256, 256, 0, stream>>>(packed);
  rnorm_kernel<<<(6 * CNT) / 8, 256, 0, stream>>>(descs, rnorm);
  if (big_ws) {
    xn_kernel<<<(6 * CNT * DIM / 4) / 256, 256, 0, stream>>>(descs, rnorm, xn);
    gemm_min_async_kernel<<<dim3(CNT / BN, CNT / BM, NMAT), 256, 0, stream>>>(xn, packed);
  } else {
    gemm_min_kernel<<<dim3(CNT / BN, CNT / BM, NMAT), 256, 0, stream>>>(descs, rnorm, packed);
  }
  posneg_kernel<<<(NMAT * CNT) / 8, 256, 0, stream>>>(descs, rnorm, packed, dots);
  alpha_stats_kernel<<<1, 256, 0, stream>>>(dots, stats);
  bce_kernel<<<200, 256, 0, stream>>>(dots, stats, bce);
  final_kernel<<<1, 256, 0, stream>>>(dots, bce, l_a, l_p, out);
}